// GCNModel_23055384445705
// MI455X (gfx1250) — compile-verified
//
#include <hip/hip_runtime.h>
#include <cstdint>

typedef __attribute__((ext_vector_type(2))) float v2f;
typedef __attribute__((ext_vector_type(8))) float v8f;

#define HSIZE (1u << 20)
#define HMASK (HSIZE - 1u)

// ---------------------------------------------------------------------------
// Edge dedup via open-addressing hash (key = min(a,b)*n + max(a,b)).
// First inserter of a key appends the undirected edge and bumps both degrees,
// exactly matching jnp.unique + segment_sum degree semantics (no self-loops
// exist in the input by construction).
// ---------------------------------------------------------------------------
__global__ void dedup_kernel(const int* __restrict__ edges, int E, int n,
                             unsigned long long* __restrict__ table,
                             uint2* __restrict__ uedges,
                             unsigned int* __restrict__ count,
                             int* __restrict__ deg)
{
    int e = blockIdx.x * blockDim.x + threadIdx.x;
    if (e >= E) return;
    int a = edges[2 * e + 0];
    int b = edges[2 * e + 1];
    unsigned u = (unsigned)min(a, b);
    unsigned v = (unsigned)max(a, b);
    unsigned long long key = (unsigned long long)u * (unsigned)n + v;
    unsigned h = (unsigned)((key * 0x9E3779B97F4A7C15ull) >> 40) & HMASK;
    const unsigned long long EMPTY = ~0ull;
    while (true) {
        unsigned long long prev = atomicCAS(&table[h], EMPTY, key);
        if (prev == EMPTY) {
            unsigned idx = atomicAdd(count, 1u);
            uedges[idx] = make_uint2(u, v);
            atomicAdd(&deg[u], 1);
            atomicAdd(&deg[v], 1);
            break;
        }
        if (prev == key) break;  // duplicate edge
        h = (h + 1u) & HMASK;
    }
}

__global__ void invdeg_kernel(const int* __restrict__ deg,
                              float* __restrict__ invdeg, int n)
{
    int i = blockIdx.x * blockDim.x + threadIdx.x;
    if (i < n) {
        int d = deg[i];
        invdeg[i] = d > 0 ? 1.0f / (float)d : 0.0f;
    }
}

// ---------------------------------------------------------------------------
// Symmetric neighbor-sum: out[u] += feat[v], out[v] += feat[u] per unique
// edge. One thread handles one (edge, 4-feature chunk); L2-resident since
// feat (<=80MB) and out fit in the 192MB global L2.
// ---------------------------------------------------------------------------
__global__ void agg_edges(const uint2* __restrict__ uedges,
                          const unsigned int* __restrict__ countPtr,
                          const float* __restrict__ feat,   // n x F
                          float* __restrict__ out,          // n x F, pre-zeroed
                          int F, int chunkShift)            // chunks = F/4 = 1<<chunkShift
{
    const unsigned int cnt = *countPtr;
    const size_t total = (size_t)cnt << chunkShift;
    const unsigned int cmask = (1u << chunkShift) - 1u;
    for (size_t wi = (size_t)blockIdx.x * blockDim.x + threadIdx.x; wi < total;
         wi += (size_t)gridDim.x * blockDim.x) {
        const size_t e = wi >> chunkShift;
        const int c = (int)(wi & cmask) * 4;
        uint2 uv = uedges[e];
        const float4 xv = *(const float4*)(feat + (size_t)uv.y * F + c);
        const float4 xu = *(const float4*)(feat + (size_t)uv.x * F + c);
        float* ou = out + (size_t)uv.x * F + c;
        float* ov = out + (size_t)uv.y * F + c;
        unsafeAtomicAdd(ou + 0, xv.x);
        unsafeAtomicAdd(ou + 1, xv.y);
        unsafeAtomicAdd(ou + 2, xv.z);
        unsafeAtomicAdd(ou + 3, xv.w);
        unsafeAtomicAdd(ov + 0, xu.x);
        unsafeAtomicAdd(ov + 1, xu.y);
        unsafeAtomicAdd(ov + 2, xu.z);
        unsafeAtomicAdd(ov + 3, xu.w);
    }
}

// ---------------------------------------------------------------------------
// C[Nrows x Nc] = diag(invdeg) * A[Nrows x K] @ B[K x Nc] using
// V_WMMA_F32_16X16X4_F32 (full f32 precision, matches reference numerics).
// One wave computes one 16x16 tile of C; K-loop in steps of 4.
// VGPR layouts per CDNA5 ISA 7.12.2:
//   A 16x4 : lane<16 holds K=k..k+1, lane>=16 holds K=k+2..k+3 (row M = lane&15)
//   B 4x16 : v0 = rows {k, k+2}, v1 = rows {k+1, k+3} split by lane half
//   C 16x16: VGPR r -> row r + 8*(lane>=16), col = lane&15
// Wave-uniform early exit keeps EXEC all-ones at every WMMA.
// ---------------------------------------------------------------------------
__global__ void gemm_scaled_wmma(const float* __restrict__ A,
                                 const float* __restrict__ invdeg,
                                 const float* __restrict__ B,
                                 float* __restrict__ C,
                                 int Nrows, int K, int Nc)
{
    const int wavesPerBlock = blockDim.x >> 5;
    const int waveId = blockIdx.x * wavesPerBlock + ((int)threadIdx.x >> 5);
    const int tilesN = Nc >> 4;
    const int tileM = waveId / tilesN;
    const int tileN = waveId % tilesN;
    if (tileM * 16 >= Nrows) return;          // wave-uniform: EXEC stays full

    const int lane = (int)threadIdx.x & 31;
    const int half = lane >> 4;               // 0: lanes 0-15, 1: lanes 16-31
    const int l16  = lane & 15;

    int arow = tileM * 16 + l16;
    if (arow >= Nrows) arow = Nrows - 1;      // safe clamp (N%16==0 here)
    const float s = invdeg[arow];
    const float* Ar = A + (size_t)arow * K;
    const int col = tileN * 16 + l16;

    v8f acc = {};
#pragma unroll 4
    for (int k = 0; k < K; k += 4) {
        const int kk = k + 2 * half;
        v2f a, b;
        a.x = Ar[kk + 0] * s;
        a.y = Ar[kk + 1] * s;
        b.x = B[(size_t)(kk + 0) * Nc + col];
        b.y = B[(size_t)(kk + 1) * Nc + col];
        acc = __builtin_amdgcn_wmma_f32_16x16x4_f32(
            /*neg_a=*/false, a, /*neg_b=*/false, b,
            /*c_mod=*/(short)0, acc, /*reuse_a=*/false, /*reuse_b=*/false);
    }

#pragma unroll
    for (int r = 0; r < 8; ++r) {
        const int orow = tileM * 16 + r + 8 * half;
        if (orow < Nrows)
            C[(size_t)orow * Nc + col] = acc[r];
    }
}

// ---------------------------------------------------------------------------
extern "C" void kernel_launch(void* const* d_in, const int* in_sizes, int n_in,
                              void* d_out, int out_size, void* d_ws, size_t ws_size,
                              hipStream_t stream)
{
    const float* x     = (const float*)d_in[0];
    const int*   edges = (const int*)d_in[1];
    const float* Wh    = (const float*)d_in[3];
    const float* Wo    = (const float*)d_in[4];

    const int N     = in_sizes[2];            // node_graph_ind has N entries
    const int E     = in_sizes[1] / 2;
    const int F_IN  = in_sizes[0] / N;        // 512
    const int F_HID = in_sizes[3] / F_IN;     // 256
    const int F_OUT = in_sizes[4] / F_HID;    // 64
    (void)F_OUT; (void)n_in; (void)ws_size; (void)out_size;

    // ---- carve workspace -------------------------------------------------
    char* ws = (char*)d_ws;
    size_t off = 0;
    auto carve = [&](size_t bytes) -> void* {
        void* p = ws + off;
        off = (off + bytes + 255) & ~(size_t)255;
        return p;
    };
    unsigned long long* table = (unsigned long long*)carve((size_t)HSIZE * 8);
    unsigned int*       count = (unsigned int*)carve(sizeof(unsigned int));
    int*                deg   = (int*)carve((size_t)N * 4);
    float*              ideg  = (float*)carve((size_t)N * 4);
    uint2*              ued   = (uint2*)carve((size_t)E * 8);
    float*              agg1  = (float*)carve((size_t)N * F_IN * 4);
    float*              hbuf  = (float*)carve((size_t)N * F_HID * 4);
    float*              agg2  = (float*)carve((size_t)N * F_HID * 4);

    // ---- init (graph-capture-safe memsets) -------------------------------
    hipMemsetAsync(table, 0xFF, (size_t)HSIZE * 8, stream);
    hipMemsetAsync(count, 0, sizeof(unsigned int), stream);
    hipMemsetAsync(deg,   0, (size_t)N * 4, stream);
    hipMemsetAsync(agg1,  0, (size_t)N * F_IN * 4, stream);
    hipMemsetAsync(agg2,  0, (size_t)N * F_HID * 4, stream);

    // ---- 1) dedup + degree ----------------------------------------------
    dedup_kernel<<<(E + 255) / 256, 256, 0, stream>>>(edges, E, N, table, ued, count, deg);

    // ---- 2) 1/deg --------------------------------------------------------
    invdeg_kernel<<<(N + 255) / 256, 256, 0, stream>>>(deg, ideg, N);

    // ---- 3) agg1 = A_sym @ x --------------------------------------------
    int shiftIn = 31 - __builtin_clz((unsigned)(F_IN / 4));   // 7
    agg_edges<<<8192, 256, 0, stream>>>(ued, count, x, agg1, F_IN, shiftIn);

    // ---- 4) h = (agg1/deg) @ W_hidden  via f32 WMMA ----------------------
    {
        int waves = ((N + 15) / 16) * (F_HID / 16);
        int blocks = (waves * 32 + 255) / 256;
        gemm_scaled_wmma<<<blocks, 256, 0, stream>>>(agg1, ideg, Wh, hbuf, N, F_IN, F_HID);
    }

    // ---- 5) agg2 = A_sym @ h --------------------------------------------
    int shiftHid = 31 - __builtin_clz((unsigned)(F_HID / 4)); // 6
    agg_edges<<<8192, 256, 0, stream>>>(ued, count, hbuf, agg2, F_HID, shiftHid);

    // ---- 6) out = (agg2/deg) @ W_out  via f32 WMMA -----------------------
    {
        int waves = ((N + 15) / 16) * (F_OUT / 16);
        int blocks = (waves * 32 + 255) / 256;
        gemm_scaled_wmma<<<blocks, 256, 0, stream>>>(agg2, ideg, Wo, (float*)d_out, N, F_HID, F_OUT);
    }
}